// FokkerPlanck2D_Base_4372276707742
// MI455X (gfx1250) — compile-verified
//
#include <hip/hip_runtime.h>

// Problem constants (from reference: B=128, Nx=Ny=512, fp32)
#define NXg 512
#define NYg 512
#define TX  8      // interior tile rows per block
#define TY  128    // interior tile cols per block
#define HTX (TX + 2)        // 10 rows incl. halo
#define HTY (TY + 2)        // 130 cols incl. halo
#define LPAD 4              // left pad so interior col starts 16B-aligned in LDS
#define LDSW (TY + 2 * LPAD) // 136 floats per LDS row (544B, 16B-aligned stride)
#define NTHREADS 256

// ---- CDNA5 async global->LDS path (guarded; sync fallback keeps compile green) ----
#if defined(__has_builtin)
#if __has_builtin(__builtin_amdgcn_global_load_async_to_lds_b128) && \
    __has_builtin(__builtin_amdgcn_global_load_async_to_lds_b32)
#define HAVE_ASYNC_LDS 1
#endif
#if __has_builtin(__builtin_amdgcn_s_wait_asynccnt)
#define HAVE_WAIT_ASYNC 1
#endif
#endif

// Typed address-space pointers matching the builtin prototypes (probe-verified):
//  b32 : (int AS1*, int AS3*, imm offset, imm cpol)
//  b128: (int4 AS1*, int4 AS3*, imm offset, imm cpol)
typedef int v4i __attribute__((ext_vector_type(4)));
typedef __attribute__((address_space(1))) int  ga_i32_t;
typedef __attribute__((address_space(3))) int  la_i32_t;
typedef __attribute__((address_space(1))) v4i  ga_v4i_t;
typedef __attribute__((address_space(3))) v4i  la_v4i_t;

__device__ __forceinline__ void async_g2l_b128(const float* g, float* l) {
#ifdef HAVE_ASYNC_LDS
  __builtin_amdgcn_global_load_async_to_lds_b128((ga_v4i_t*)g, (la_v4i_t*)l, 0, 0);
#else
  *(float4*)l = *(const float4*)g;
#endif
}

__device__ __forceinline__ void async_g2l_b32(const float* g, float* l) {
#ifdef HAVE_ASYNC_LDS
  __builtin_amdgcn_global_load_async_to_lds_b32((ga_i32_t*)g, (la_i32_t*)l, 0, 0);
#else
  *l = *g;
#endif
}

__device__ __forceinline__ void wait_async_lds() {
#ifdef HAVE_WAIT_ASYNC
  __builtin_amdgcn_s_wait_asynccnt(0);
#elif defined(HAVE_ASYNC_LDS)
  asm volatile("s_wait_asynccnt 0x0" ::: "memory");
#endif
}

__global__ __launch_bounds__(NTHREADS)
void fp2d_step_kernel(const float* __restrict__ f,
                      const float* __restrict__ A,
                      const float* __restrict__ D,
                      const float* __restrict__ dtp,
                      float* __restrict__ out)
{
  __shared__ float sF  [HTX * LDSW];
  __shared__ float sFx [HTX * LDSW];
  __shared__ float sFy [HTX * LDSW];
  __shared__ float sGxx[HTX * LDSW];
  __shared__ float sGyy[HTX * LDSW];
  __shared__ float sGxy[HTX * LDSW];

  const int tid = threadIdx.x;
  const int j0  = blockIdx.x * TY;      // tile origin (interior) in j
  const int i0  = blockIdx.y * TX;      // tile origin (interior) in i
  const int b   = blockIdx.z;

  const size_t plane = (size_t)NXg * NYg;
  const float* fb = f + (size_t)b * plane;
  const float* A0 = A;
  const float* A1 = A + plane;
  const float* D0 = D;
  const float* D1 = D + plane;
  const float* D2 = D + 2 * plane;

  // ---- Phase 1: load-or-zero the f halo tile. Async global->LDS writes and ds
  // zero-stores hit DISJOINT LDS addresses (each element is exactly one of the
  // two), so no pre-zero pass and no extra barrier are needed.
  const float4 zero4 = make_float4(0.0f, 0.0f, 0.0f, 0.0f);

  // Bulk: b128 over interior columns (always in range in j); zero OOB rows.
  for (int idx = tid; idx < HTX * (TY / 4); idx += NTHREADS) {
    const int li = idx >> 5;            // TY/4 == 32
    const int q  = idx & 31;
    const int gi = i0 - 1 + li;
    const int c  = li * LDSW + LPAD + 4 * q;
    if (gi >= 0 && gi < NXg)
      async_g2l_b128(fb + (size_t)gi * NYg + j0 + 4 * q, &sF[c]);
    else
      *(float4*)&sF[c] = zero4;
  }
  // Halo columns (j0-1 and j0+TY), b32; zero where out of grid.
  for (int idx = tid; idx < HTX * 2; idx += NTHREADS) {
    const int li   = idx >> 1;
    const int side = idx & 1;
    const int gi   = i0 - 1 + li;
    const int gj   = side ? (j0 + TY) : (j0 - 1);
    const int c    = li * LDSW + (side ? (LPAD + TY) : (LPAD - 1));
    if (gi >= 0 && gi < NXg && gj >= 0 && gj < NYg)
      async_g2l_b32(fb + (size_t)gi * NYg + gj, &sF[c]);
    else
      sF[c] = 0.0f;
  }

  // ---- Prefetch coefficient tile rows while async f-loads are in flight
  // (A/D are 5 MB total -> L2-resident, reused by all 128 batch slices).
  // 5 planes x 10 rows x 5 cachelines (128B) = 250 prefetches.
  if (tid < 5 * HTX * 5) {
    const int p  = tid / (HTX * 5);
    const int r  = (tid / 5) % HTX;
    const int ch = tid % 5;
    const int gi = min(max(i0 - 1 + r, 0), NXg - 1);
    const int gj = min(max(j0 - 1 + ch * 32, 0), NYg - 1);
    const float* pl = (p == 0) ? A0 : (p == 1) ? A1 : (p == 2) ? D0 : (p == 3) ? D1 : D2;
    __builtin_prefetch(pl + (size_t)gi * NYg + gj, 0, 3);
  }

  wait_async_lds();     // per-wave: ASYNCcnt -> 0 (LDS writes landed)
  __syncthreads();      // cross-wave visibility (also covers the ds zero-stores)

  const float dt = *dtp;

  // ---- Phase 2: product fields Fx=A0*f, Fy=A1*f, Gxx=D0*f, Gyy=D1*f, Gxy=D2*f ----
  // Bulk: b128 coefficient loads over interior columns (gi clamped; f halo is 0 so
  // clamped-coefficient * 0 == 0 matches the zero-padded reference).
  for (int idx = tid; idx < HTX * (TY / 4); idx += NTHREADS) {
    const int li  = idx >> 5;
    const int q   = idx & 31;
    const int gic = min(max(i0 - 1 + li, 0), NXg - 1);
    const size_t g = (size_t)gic * NYg + j0 + 4 * q;
    const int c = li * LDSW + LPAD + 4 * q;

    const float4 fv = *(const float4*)&sF[c];
    const float4 a0 = *(const float4*)(A0 + g);
    const float4 a1 = *(const float4*)(A1 + g);
    const float4 d0 = *(const float4*)(D0 + g);
    const float4 d1 = *(const float4*)(D1 + g);
    const float4 d2 = *(const float4*)(D2 + g);

    float4 t;
    t.x = a0.x * fv.x; t.y = a0.y * fv.y; t.z = a0.z * fv.z; t.w = a0.w * fv.w;
    *(float4*)&sFx[c] = t;
    t.x = a1.x * fv.x; t.y = a1.y * fv.y; t.z = a1.z * fv.z; t.w = a1.w * fv.w;
    *(float4*)&sFy[c] = t;
    t.x = d0.x * fv.x; t.y = d0.y * fv.y; t.z = d0.z * fv.z; t.w = d0.w * fv.w;
    *(float4*)&sGxx[c] = t;
    t.x = d1.x * fv.x; t.y = d1.y * fv.y; t.z = d1.z * fv.z; t.w = d1.w * fv.w;
    *(float4*)&sGyy[c] = t;
    t.x = d2.x * fv.x; t.y = d2.y * fv.y; t.z = d2.z * fv.z; t.w = d2.w * fv.w;
    *(float4*)&sGxy[c] = t;
  }
  // Edge columns of the halo tile (scalar, both indices clamped)
  for (int idx = tid; idx < HTX * 2; idx += NTHREADS) {
    const int li   = idx >> 1;
    const int side = idx & 1;
    const int gic  = min(max(i0 - 1 + li, 0), NXg - 1);
    const int gjc  = min(max(side ? (j0 + TY) : (j0 - 1), 0), NYg - 1);
    const size_t g = (size_t)gic * NYg + gjc;
    const int c = li * LDSW + (side ? (LPAD + TY) : (LPAD - 1));
    const float fv = sF[c];
    sFx [c] = A0[g] * fv;
    sFy [c] = A1[g] * fv;
    sGxx[c] = D0[g] * fv;
    sGyy[c] = D1[g] * fv;
    sGxy[c] = D2[g] * fv;
  }
  __syncthreads();

  // ---- Phase 3: 9-point stencil; each thread -> 4 consecutive j, one b128 store ----
  const int li  = 1 + (tid >> 5);                 // 1..8  (interior row in tile)
  const int ljb = LPAD + ((tid & 31) << 2);       // interior col start in LDS
  const int c0  = li * LDSW + ljb;

  float res[4];
#pragma unroll
  for (int k = 0; k < 4; ++k) {
    const int cc = c0 + k;
    const float dFx   = 0.5f * (sFx[cc + LDSW] - sFx[cc - LDSW]);
    const float dFy   = 0.5f * (sFy[cc + 1] - sFy[cc - 1]);
    const float d2Gxx = sGxx[cc + LDSW] - 2.0f * sGxx[cc] + sGxx[cc - LDSW];
    const float d2Gyy = sGyy[cc + 1]    - 2.0f * sGyy[cc] + sGyy[cc - 1];
    const float d2Gxy = 0.25f * (sGxy[cc + LDSW + 1] - sGxy[cc + LDSW - 1]
                               - sGxy[cc - LDSW + 1] + sGxy[cc - LDSW - 1]);
    const float rhs = -(dFx + dFy) + 0.5f * (d2Gxx + d2Gyy) + d2Gxy;
    res[k] = fmaxf(fmaf(dt, rhs, sF[cc]), 0.0f);
  }

  float4 o;
  o.x = res[0]; o.y = res[1]; o.z = res[2]; o.w = res[3];
  *(float4*)(out + (size_t)b * plane + (size_t)(i0 + li - 1) * NYg + (j0 + (ljb - LPAD))) = o;
}

extern "C" void kernel_launch(void* const* d_in, const int* in_sizes, int n_in,
                              void* d_out, int out_size, void* d_ws, size_t ws_size,
                              hipStream_t stream) {
  const float* f  = (const float*)d_in[0];
  const float* A  = (const float*)d_in[1];
  const float* D  = (const float*)d_in[2];
  const float* dt = (const float*)d_in[3];
  float* out = (float*)d_out;

  const int B = in_sizes[0] / (NXg * NYg);   // 128
  dim3 grid(NYg / TY, NXg / TX, B);          // (4, 64, 128)
  fp2d_step_kernel<<<grid, NTHREADS, 0, stream>>>(f, A, D, dt, out);
}